// WrappedNormal_6983616824104
// MI455X (gfx1250) — compile-verified
//
#include <hip/hip_runtime.h>
#include <hip/hip_bf16.h>

#define N_NODES 100000
#define ROWLEN  17            // table row: x0 + 16 spatial dims
#define LOG2F_C 0.69314718055994530942f
#define LOG_2PI_D 1.8378770664093454836
#define INV_NM1 (1.0f / 99999.0f)   // 1/(N_NODES-1)

typedef __attribute__((ext_vector_type(2))) float v2f;
typedef __attribute__((ext_vector_type(8))) float v8f;

// ---------------------------------------------------------------------------
// Kernel 1: invert Sigma (16x16 SPD) + log-det constant. Runs once, 1 thread.
// ws[0..255] = Sigma^-1 (row major), ws[256] = const term.
// ---------------------------------------------------------------------------
__global__ void sigma_prep_kernel(const float* __restrict__ Sigma,
                                  float* __restrict__ ws) {
    if (threadIdx.x != 0 || blockIdx.x != 0) return;
    double M[16][32];
    for (int i = 0; i < 16; ++i)
        for (int j = 0; j < 16; ++j) {
            M[i][j]      = (double)Sigma[i * 16 + j];
            M[i][16 + j] = (i == j) ? 1.0 : 0.0;
        }
    double det = 1.0;
    for (int k = 0; k < 16; ++k) {            // SPD, well-conditioned: no pivoting
        double p = M[k][k];
        det *= p;
        double pinv = 1.0 / p;
        for (int j = 0; j < 32; ++j) M[k][j] *= pinv;
        for (int i = 0; i < 16; ++i) {
            if (i == k) continue;
            double f = M[i][k];
            for (int j = 0; j < 32; ++j) M[i][j] -= f * M[k][j];
        }
    }
    for (int i = 0; i < 16; ++i)
        for (int j = 0; j < 16; ++j)
            ws[i * 16 + j] = (float)M[i][16 + j];
    ws[256] = (float)(8.0 * LOG_2PI_D + 0.5 * log(det + 1e-6));
}

// ---------------------------------------------------------------------------
// Kernel 2: per-node latent likelihood. 16 nodes per wave; maha via f32 WMMA:
// D = V(16x16) @ SigmaInv(16x16) using 4x v_wmma_f32_16x16x4_f32 (K chunks).
// ---------------------------------------------------------------------------
__global__ void node_lik_kernel(const float* __restrict__ table,
                                const float* __restrict__ sinv,
                                const float* __restrict__ cconst_p,
                                float* __restrict__ lik) {
    __shared__ float Vt[8][16 * 17];   // per-wave v tiles, row stride 17 (bank-conflict free)
    __shared__ float Dt[8][16 * 17];   // per-wave D = V @ P tiles

    const int lane = threadIdx.x & 31;
    const int wib  = threadIdx.x >> 5;
    const int wave = blockIdx.x * (blockDim.x >> 5) + wib;
    const int base = wave * 16;
    const bool active = (base < N_NODES);   // uniform per wave
    const int m16 = lane & 15;
    const int kh  = lane >> 4;              // lane half: selects K pairs per ISA A-layout
    const int node = base + m16;

    float* Vw = Vt[wib];
    float* Dw = Dt[wib];

    if (active) {
        const float* row = table + (long long)node * ROWLEN;
        float x0    = row[0];
        float alpha = fmaxf(x0, 1.0f + 1e-10f);
        float coef  = acoshf(alpha) / sqrtf(alpha * alpha - 1.0f);
        const int j0 = kh * 8;             // lanes 0-15 write dims 0..7, 16-31 dims 8..15
        #pragma unroll
        for (int j = 0; j < 8; ++j)
            Vw[m16 * 17 + j0 + j] = coef * row[1 + j0 + j];
    }
    __syncthreads();

    // A layout (16x4 f32): lane M = lane%16; VGPR0/1 hold K = 2*(lane/16)+{0,1}
    // B layout (4x16 f32): lane N = lane%16; VGPR0/1 hold K = 2*(lane/16)+{0,1}
    v8f acc = {};
    #pragma unroll
    for (int c = 0; c < 4; ++c) {
        const int d0 = 4 * c + 2 * kh;
        v2f a, b;
        a.x = Vw[m16 * 17 + d0];
        a.y = Vw[m16 * 17 + d0 + 1];
        b.x = sinv[d0 * 16 + m16];
        b.y = sinv[(d0 + 1) * 16 + m16];
        acc = __builtin_amdgcn_wmma_f32_16x16x4_f32(
            /*neg_a=*/false, a, /*neg_b=*/false, b,
            /*c_mod=*/(short)0, acc, /*reuse_a=*/false, /*reuse_b=*/false);
    }

    // C/D layout: VGPR r holds row M=r (lanes 0-15) / M=r+8 (lanes 16-31), N=lane%16
    #pragma unroll
    for (int r = 0; r < 8; ++r)
        Dw[(r + 8 * kh) * 17 + m16] = acc[r];
    __syncthreads();

    if (active && lane < 16) {
        float maha = 0.0f, vnsq = 0.0f;
        #pragma unroll
        for (int n = 0; n < 16; ++n) {
            float vv = Vw[m16 * 17 + n];
            maha += Dw[m16 * 17 + n] * vv;
            vnsq += vv * vv;
        }
        maha *= 0.5f;
        float vn = sqrtf(vnsq);
        if (vn <= 1e-6f) vn = 1e-6f;
        float tang = 15.0f * (log1pf(-expf(-2.0f * vn)) + vn - LOG2F_C - logf(vn));
        lik[node] = *cconst_p + maha + tang;
    }
}

// ---------------------------------------------------------------------------
// Kernel 3: per-pair loss. Bandwidth-bound gather kernel; table + lik are
// L2-resident (7.2 MB << 192 MB), so HBM traffic is pairs/labels/out only.
// ---------------------------------------------------------------------------
__global__ void pair_loss_kernel(const long long* __restrict__ pairs,
                                 const long long* __restrict__ labels,
                                 const float* __restrict__ table,
                                 const float* __restrict__ beta,
                                 const float* __restrict__ lik,
                                 float* __restrict__ out, int nB) {
    int t = blockIdx.x * blockDim.x + threadIdx.x;
    if (t >= nB) return;
    long long i = pairs[2 * t];
    long long j = pairs[2 * t + 1];
    const float* u = table + i * ROWLEN;
    const float* v = table + j * ROWLEN;
    float u0 = u[0], v0 = v[0];
    float sp = 0.0f;
    #pragma unroll
    for (int k = 1; k < ROWLEN; ++k) sp = fmaf(u[k], v[k], sp);
    float alpha = fmaxf(u0 * v0 - sp, 1.0f + 1e-10f);   // -lorentz_inner, clipped
    float dist  = acoshf(alpha);
    float s = beta[0] * (dist - 10.0f);
    float x = (labels[t] == 1) ? s : -s;
    float loss = fmaxf(x, 0.0f) + log1pf(expf(-fabsf(x)));   // logaddexp(0, x)
    loss += (lik[i] + lik[j]) * INV_NM1;
    out[t] = loss;
}

// ---------------------------------------------------------------------------
extern "C" void kernel_launch(void* const* d_in, const int* in_sizes, int n_in,
                              void* d_out, int out_size, void* d_ws, size_t ws_size,
                              hipStream_t stream) {
    const long long* pairs  = (const long long*)d_in[0];   // int64 (B,2)
    const long long* labels = (const long long*)d_in[1];   // int64 (B,)
    const float*     table  = (const float*)d_in[2];       // f32 (100000,17)
    const float*     beta   = (const float*)d_in[3];       // f32 (1,)
    const float*     Sigma  = (const float*)d_in[4];       // f32 (16,16)
    float* out = (float*)d_out;
    float* ws  = (float*)d_ws;
    float* sinv   = ws;          // 256 floats
    float* cconst = ws + 256;    // 1 float
    float* lik    = ws + 320;    // N_NODES floats

    const int B = out_size;

    hipLaunchKernelGGL(sigma_prep_kernel, dim3(1), dim3(1), 0, stream, Sigma, ws);

    const int waves  = (N_NODES + 15) / 16;       // 6250 (N divides by 16 exactly)
    const int blocks = (waves + 7) / 8;           // 8 waves / 256-thread block
    hipLaunchKernelGGL(node_lik_kernel, dim3(blocks), dim3(256), 0, stream,
                       table, sinv, cconst, lik);

    hipLaunchKernelGGL(pair_loss_kernel, dim3((B + 255) / 256), dim3(256), 0, stream,
                       pairs, labels, table, beta, lik, out, B);
}